// PEPS_47545287967499
// MI455X (gfx1250) — compile-verified
//
#include <hip/hip_runtime.h>
#include <hip/hip_bf16.h>
#include <math.h>

// PEPS boundary-MPS contraction with zip-up SVD truncation, MI455X (gfx1250).
// One workgroup (8 wave32s) per sample. Column-major 216x216 f32 working
// matrix in LDS (needs CDNA5's 320KB WGP LDS). Carry = U^T @ mat computed
// with v_wmma_f32_16x16x4_f32 over zero-padded, pre-packed operands so the
// K-loop is branch-free (ds_load_b64 + global_load_b64 + wmma) and the
// store-back is unconditional (padded carry buffer).
// Workspace need: 32 * 169792 * 4 B ~= 21.7 MB.

#define R_ 8
#define C_ 8
#define Dd 6
#define CHI 36
#define LDW 216          // leading dim of LDS working matrix (column-major)
#define LDT 224          // leading dim (rows, padded) of transposed buffers
#define MAXB 216

typedef __attribute__((ext_vector_type(2))) float v2f;
typedef __attribute__((ext_vector_type(8))) float v8f;

// per-sample workspace layout (floats)
#define MPS_SITE 7776            // 36*6*36 padded (l stride 216, p stride 36)
#define MPS_OFF  0               // 8 * 7776 = 62208
#define A_OFF    62208           // 46656
#define MO_OFF   108864          // matOT: 224*224 = 50176 (col-major, padded)
#define CAR_OFF  159040          // carry: 48*224 = 10752 (padded LD)
#define WS_PER   169792

__global__ __launch_bounds__(256, 1)
void peps_logamp_kernel(const float* __restrict__ tensors,
                        const int* __restrict__ x,
                        float* __restrict__ out,
                        float* __restrict__ ws)
{
    __shared__ __align__(16) float W[MAXB * LDW];   // 186.6 KB, column-major
    __shared__ __align__(16) float UsT[48 * LDT];   // 43 KB: packed U*invS, T
    __shared__ float w_s[Dd * Dd * Dd * Dd];
    __shared__ float norms[MAXB];
    __shared__ float invS[CHI];
    __shared__ int   idxA[CHI];
    __shared__ int   used[MAXB];
    __shared__ float vecA[64], vecB[64];

    const int s    = blockIdx.x;
    const int tid  = threadIdx.x;
    const int lane = tid & 31;
    const int wave = tid >> 5;

    float* base  = ws + (size_t)s * WS_PER;
    float* mps   = base + MPS_OFF;
    float* Abuf  = base + A_OFF;
    float* matOT = base + MO_OFF;     // [colP][rowP], LD = LDT, zero-padded
    float* carry = base + CAR_OFF;    // [row][col], LD = carLD (padded)

    const int* spins = x + s * (R_ * C_);

    // boundary MPS = 8 trivial (1,1,1) tensors
    int lo[8], po[8], ro[8];
    for (int c = 0; c < 8; ++c) { lo[c] = 1; po[c] = 1; ro[c] = 1; }
    for (int c = tid; c < 8; c += 256) mps[c * MPS_SITE] = 1.0f;
    __syncthreads();

    for (int r = 0; r < R_; ++r) {
        int Kc = 1, carDl = 1, carWl = 1, carLD = 1;   // carry dims + pad LD
        for (int c = 0; c < C_; ++c) {
            const int up    = (r == 0)      ? 1 : Dd;
            const int down  = (r == R_ - 1) ? 1 : Dd;
            const int left  = (c == 0)      ? 1 : Dd;
            const int right = (c == C_ - 1) ? 1 : Dd;
            const int spin  = spins[r * C_ + c];
            const float* tsel = tensors + ((size_t)r * 20736 + (size_t)c * 2592
                                           + (size_t)spin * 1296);
            if (lane == 0) __builtin_prefetch(tsel, 0, 1);   // global_prefetch_b8

            // MPO tensor w[l][rb][u][d] = tsel[u][d][l][rb]
            const int wtot = left * right * up * down;
            for (int e = tid; e < wtot; e += 256) {
                int d = e % down; int tmp = e / down;
                int u = tmp % up; tmp /= up;
                int rb = tmp % right; int l = tmp / right;
                w_s[((l * right + rb) * up + u) * down + d] =
                    tsel[u * 216 + d * 36 + l * 6 + rb];
            }
            if (c == 0) {
                if (tid == 0) carry[0] = 1.0f;      // identity carry (1,1,1)
                Kc = 1; carDl = lo[0]; carWl = left; carLD = 1;
            }
            __syncthreads();

            const int Dl = lo[c], Dr = ro[c];
            const float* m = mps + c * MPS_SITE;

            // step 1: A[kk][l][u][dr] = sum_dl carry[kk][dl][l] * m[dl][u][dr]
            const int atot = Kc * left * up * Dr;
            for (int e = tid; e < atot; e += 256) {
                int dr = e % Dr; int tmp = e / Dr;
                int u = tmp % up; tmp /= up;
                int l = tmp % left; int kk = tmp / left;
                float acc = 0.f;
                for (int dl = 0; dl < Dl; ++dl)
                    acc += carry[kk * carLD + dl * carWl + l]
                         * m[dl * 216 + u * 36 + dr];
                Abuf[((kk * left + l) * up + u) * Dr + dr] = acc;
            }
            __syncthreads();

            // step 2: theta (mrows x ncols) + 1e-7*I.
            // W column-major (LDS); matOT transposed+zero-padded (global).
            const int mrows  = Kc * down, ncols = Dr * right;
            const int mrowsP = (mrows + 15) & ~15;
            const int ncolsP = (ncols + 15) & ~15;
            const int mtotP  = mrowsP * ncolsP;
            for (int e = tid; e < mtotP; e += 256) {
                int row = e % mrowsP, col = e / mrowsP;
                float acc = 0.f;
                if (row < mrows && col < ncols) {
                    int rb = col % right, dr = col / right;
                    int d  = row % down,  kk = row / down;
                    acc = (row == col) ? 1e-7f : 0.f;
                    for (int l = 0; l < left; ++l)
                        for (int u = 0; u < up; ++u)
                            acc += Abuf[((kk * left + l) * up + u) * Dr + dr]
                                 * w_s[((l * right + rb) * up + u) * down + d];
                    W[col * LDW + row] = acc;
                }
                matOT[col * LDT + row] = acc;
            }
            __syncthreads();

            if (c == C_ - 1) {   // last site: no SVD, keep full theta
                const int mtot = mrows * ncols;
                for (int e = tid; e < mtot; e += 256) {
                    int col = e % ncols, row = e / ncols;
                    int d = row % down, kk = row / down;
                    mps[7 * MPS_SITE + kk * 216 + d * 36 + col] =
                        W[col * LDW + row];
                }
                lo[7] = Kc; po[7] = down; ro[7] = ncols;
                __syncthreads();
                break;
            }

            // step 3: one-sided Jacobi SVD on W columns (-> U*S).
            // Column-major: per-lane accesses are contiguous (bank friendly).
            const int npad = ncols + (ncols & 1);
            for (int sweep = 0; sweep < 8; ++sweep) {
                for (int round = 0; round < npad - 1; ++round) {
                    for (int pi = wave; pi < npad / 2; pi += 8) {
                        int a, b;
                        if (pi == 0) { a = npad - 1; b = round % (npad - 1); }
                        else {
                            a = (round + pi) % (npad - 1);
                            b = (round - pi + (npad - 1)) % (npad - 1);
                        }
                        if (a < ncols && b < ncols) {
                            float* Wa = W + a * LDW;
                            float* Wb = W + b * LDW;
                            float app = 0.f, aqq = 0.f, apq = 0.f;
                            for (int rr = lane; rr < mrows; rr += 32) {
                                float wa = Wa[rr], wb = Wb[rr];
                                app += wa * wa; aqq += wb * wb; apq += wa * wb;
                            }
                            for (int mk = 16; mk >= 1; mk >>= 1) {
                                app += __shfl_xor(app, mk, 32);
                                aqq += __shfl_xor(aqq, mk, 32);
                                apq += __shfl_xor(apq, mk, 32);
                            }
                            if (apq * apq > 1e-30f * app * aqq + 1e-37f) {
                                float tau = (aqq - app) / (2.f * apq);
                                float sg  = (tau >= 0.f) ? 1.f : -1.f;
                                float tt  = sg / (fabsf(tau) + sqrtf(1.f + tau * tau));
                                float cth = rsqrtf(1.f + tt * tt);
                                float sth = tt * cth;
                                for (int rr = lane; rr < mrows; rr += 32) {
                                    float wa = Wa[rr], wb = Wb[rr];
                                    Wa[rr] = cth * wa - sth * wb;
                                    Wb[rr] = sth * wa + cth * wb;
                                }
                            }
                        }
                    }
                    __syncthreads();
                }
            }

            // step 4: singular values = column norms; pick top-k columns
            for (int col = tid; col < ncols; col += 256) {
                float acc = 0.f;
                for (int rr = 0; rr < mrows; ++rr) {
                    float wv = W[col * LDW + rr]; acc += wv * wv;
                }
                norms[col] = sqrtf(acc);
                used[col]  = 0;
            }
            __syncthreads();
            const int kkeep = min(CHI, min(mrows, ncols));
            if (tid == 0) {
                for (int j = 0; j < kkeep; ++j) {
                    int best = 0; float bv = -1.f;
                    for (int col = 0; col < ncols; ++col)
                        if (!used[col] && norms[col] > bv) { bv = norms[col]; best = col; }
                    used[best] = 1; idxA[j] = best;
                    invS[j] = 1.f / fmaxf(bv, 1e-30f);
                }
            }
            __syncthreads();

            // step 5a: new mps[c] = U[:, top-k]   shape (Kc, down, kkeep)
            const int utot = Kc * down * kkeep;
            for (int e = tid; e < utot; e += 256) {
                int j = e % kkeep; int tmp = e / kkeep;
                int d = tmp % down; int kk = tmp / down;
                mps[c * MPS_SITE + kk * 216 + d * 36 + j] =
                    W[idxA[j] * LDW + (kk * down + d)] * invS[j];
            }
            // step 5b: pack UsT[j][row] = U[row, idxA[j]]*invS[j], zero-padded
            const int ptot = 48 * mrowsP;
            for (int e = tid; e < ptot; e += 256) {
                int row = e % mrowsP, j = e / mrowsP;
                float v = 0.f;
                if (j < kkeep && row < mrows)
                    v = W[idxA[j] * LDW + row] * invS[j];
                UsT[j * LDT + row] = v;
            }
            __syncthreads();

            // step 6: carry = S*Vh = U^T @ mat via v_wmma_f32_16x16x4_f32.
            // Branch-free K-loop: ds_load_b64 (A) + global_load_b64 (B) + wmma.
            // Unconditional store-back into padded carry (LD = ncolsP).
            const int Mt = (kkeep + 15) >> 4, Nt = (ncols + 15) >> 4;
            for (int t2 = wave; t2 < Mt * Nt; t2 += 8) {
                const int mt = t2 / Nt, nt = t2 % Nt;
                v8f acc = {};
                const int Mi  = lane & 15;
                const int hi  = lane >> 4;
                const int hi2 = hi * 2;              // K offset 0 or 2
                const int jj  = mt * 16 + Mi;        // A-frag M index (< 48)
                const int nn  = nt * 16 + Mi;        // B-frag N index (< ncolsP)
                const float* aptr = UsT   + jj * LDT + hi2;
                const float* bptr = matOT + nn * LDT + hi2;
                for (int kk0 = 0; kk0 < mrowsP; kk0 += 4) {
                    v2f af = *(const v2f*)(aptr + kk0);
                    v2f bf = *(const v2f*)(bptr + kk0);
                    acc = __builtin_amdgcn_wmma_f32_16x16x4_f32(
                              false, af, false, bf, (short)0, acc, false, false);
                }
                const int colw = nt * 16 + Mi;
                float* cptr = carry + (mt * 16 + hi * 8) * ncolsP + colw;
                #pragma unroll
                for (int i2 = 0; i2 < 8; ++i2)
                    cptr[i2 * ncolsP] = acc[i2];
            }
            __syncthreads();

            lo[c] = Kc; po[c] = down; ro[c] = kkeep;
            Kc = kkeep; carDl = Dr; carWl = right; carLD = ncolsP;
        }
        __syncthreads();
    }

    // final chain: state = prod_c mps[c][:,0,:]
    if (tid < 64) vecA[tid] = (tid == 0) ? 1.f : 0.f;
    __syncthreads();
    int len = 1;
    for (int c = 0; c < 8; ++c) {
        const int rn = ro[c];
        if (tid < 64) {
            float acc = 0.f;
            if (tid < rn)
                for (int l = 0; l < len; ++l)
                    acc += vecA[l] * mps[c * MPS_SITE + l * 216 + tid];
            vecB[tid] = acc;
        }
        __syncthreads();
        if (tid < 64) vecA[tid] = vecB[tid];
        len = rn;
        __syncthreads();
    }
    if (tid == 0) out[s] = logf(vecA[0]);
}

extern "C" void kernel_launch(void* const* d_in, const int* in_sizes, int n_in,
                              void* d_out, int out_size, void* d_ws, size_t ws_size,
                              hipStream_t stream) {
    const float* tensors = (const float*)d_in[0];   // (8,8,2,6,6,6,6) f32
    const int*   x       = (const int*)d_in[1];     // (32,64) i32
    float* out = (float*)d_out;                     // 32 f32 (log amps)
    float* ws  = (float*)d_ws;

    const int nsamples = in_sizes[1] / (R_ * C_);   // 32
    hipLaunchKernelGGL(peps_logamp_kernel, dim3(nsamples), dim3(256), 0, stream,
                       tensors, x, out, ws);
}